// Deform_Conv_48095043780730
// MI455X (gfx1250) — compile-verified
//
#include <hip/hip_runtime.h>
#include <hip/hip_bf16.h>

typedef __attribute__((ext_vector_type(2))) float v2f;
typedef __attribute__((ext_vector_type(8))) float v8f;

#define B_SZ   4
#define C_SZ   64
#define H_SZ   128
#define W_SZ   128
#define O_SZ   64
#define KK     9          // 3x3 taps
#define KDIM   (C_SZ*KK)  // 576 contraction length
#define HW     (H_SZ*W_SZ)
#define TILE_N 16

// ---------------------------------------------------------------------------
// Kernel A: offset = conv3x3(x, w_offset), pad=1, stride=1.
// One thread per output element of (B,18,H,W). 576 fp32 FMAs each; x is
// L2-resident (16.8 MB), so this is L2-bandwidth bound and cheap.
// ---------------------------------------------------------------------------
__global__ void offset_conv_kernel(const float* __restrict__ x,
                                   const float* __restrict__ w_off,
                                   float* __restrict__ off) {
    int idx = blockIdx.x * blockDim.x + threadIdx.x;
    if (idx >= B_SZ * 18 * HW) return;
    int wx = idx & (W_SZ - 1);
    int hy = (idx >> 7) & (H_SZ - 1);
    int co = (idx >> 14) % 18;
    int b  = (idx >> 14) / 18;

    const float* xb = x + (size_t)b * C_SZ * HW;
    const float* wp = w_off + co * KDIM;
    float acc = 0.f;
    for (int c = 0; c < C_SZ; ++c) {
        const float* xc = xb + c * HW;
        const float* wc = wp + c * KK;
        #pragma unroll
        for (int ky = 0; ky < 3; ++ky) {
            int yy = hy + ky - 1;
            if (yy < 0 || yy >= H_SZ) continue;
            #pragma unroll
            for (int kx = 0; kx < 3; ++kx) {
                int xx = wx + kx - 1;
                if (xx < 0 || xx >= W_SZ) continue;
                acc = fmaf(xc[yy * W_SZ + xx], wc[ky * 3 + kx], acc);
            }
        }
    }
    off[idx] = acc;
}

// ---------------------------------------------------------------------------
// Kernel B: fused bilinear gather -> LDS cols tile -> fp32 WMMA GEMM.
// Block = 128 threads (4 wave32). Each block handles 16 consecutive
// flattened positions n (all within one batch image since 16 | 16384) and
// all 64 output channels: 4 waves x one 16x16 tile each, K-loop of 144
// v_wmma_f32_16x16x4_f32 steps.
// ---------------------------------------------------------------------------
__global__ void __launch_bounds__(128)
deform_gemm_kernel(const float* __restrict__ x,
                   const float* __restrict__ off,
                   const float* __restrict__ wd,   // (O, C*9) after reshape
                   float* __restrict__ out) {
    __shared__ float colsLDS[KDIM * TILE_N];   // B-matrix tile: [kc][n], 36 KB
    __shared__ int   y0A[TILE_N * KK];
    __shared__ int   x0A[TILE_N * KK];
    __shared__ float wgtA[TILE_N * KK][4];

    const int tid  = threadIdx.x;
    const int n0   = blockIdx.x * TILE_N;      // flattened (b, h, w)
    const int b    = n0 >> 14;                 // /16384
    const int sp0  = n0 & (HW - 1);            // spatial base within image

    // ---- Phase 1: per (n_local, kk) bilinear coefficients ----
    for (int j = tid; j < TILE_N * KK; j += blockDim.x) {
        int n_l = j & (TILE_N - 1);
        int kk  = j >> 4;                      // j / 16
        int n   = n0 + n_l;
        int sp  = n & (HW - 1);
        int hy  = sp >> 7;
        int wx  = sp & (W_SZ - 1);
        // offset channels: c = kk*2 + {0:dy, 1:dx}
        const float* ob = off + ((size_t)b * 18) * HW;
        float dy = ob[(kk * 2 + 0) * HW + sp];
        float dx = ob[(kk * 2 + 1) * HW + sp];
        float py = (float)(hy - 1 + kk / 3) + dy;
        float px = (float)(wx - 1 + kk % 3) + dx;
        float y0f = floorf(py);
        float x0f = floorf(px);
        float ly = py - y0f;
        float lx = px - x0f;
        y0A[j] = (int)y0f;
        x0A[j] = (int)x0f;
        wgtA[j][0] = (1.f - ly) * (1.f - lx);
        wgtA[j][1] = (1.f - ly) * lx;
        wgtA[j][2] = ly * (1.f - lx);
        wgtA[j][3] = ly * lx;
    }
    __syncthreads();

    // ---- Phase 2: gather 64ch x 9taps x 16pos samples into LDS ----
    // s -> (c, kk, n_l); consecutive threads share (c,kk), vary n -> coalesced
    // LDS writes and near-coalesced global gathers.
    const float* xb = x + (size_t)b * C_SZ * HW;
    for (int s = tid; s < C_SZ * KK * TILE_N; s += blockDim.x) {
        int n_l = s & (TILE_N - 1);
        int kk  = (s >> 4) % KK;
        int c   = s / (KK * TILE_N);
        int j   = kk * TILE_N + n_l;
        int y0 = y0A[j], x0 = x0A[j];
        const float* xc = xb + c * HW;
        float v = 0.f;
        #pragma unroll
        for (int corner = 0; corner < 4; ++corner) {
            int yy = y0 + (corner >> 1);
            int xx = x0 + (corner & 1);
            float wgt = wgtA[j][corner];
            if (yy >= 0 && yy < H_SZ && xx >= 0 && xx < W_SZ)
                v = fmaf(xc[yy * W_SZ + xx], wgt, v);
        }
        int kc = c * KK + kk;                  // contraction index (c,k) order
        colsLDS[kc * TILE_N + n_l] = v;
    }
    __syncthreads();

    // ---- Phase 3: 16x16 output tile per wave, fp32 WMMA over K=576 ----
    const int lane = tid & 31;
    const int wave = tid >> 5;
    const int m0   = wave * 16;                // output-channel base
    const int nl   = lane & 15;
    const int hiHalf = lane >> 4;              // 0: lanes 0-15, 1: lanes 16-31

    v8f acc = {};
    const float* wrow = wd + (size_t)(m0 + nl) * KDIM;
    for (int kc0 = 0; kc0 < KDIM; kc0 += 4) {
        // A (16x4): lanes 0-15 hold K={kc0,kc0+1}, lanes 16-31 K={kc0+2,kc0+3}
        int ka = kc0 + (hiHalf << 1);
        v2f a;
        a.x = wrow[ka];
        a.y = wrow[ka + 1];
        // B (4x16): mirrored K striping, N across lanes, rows from LDS
        v2f bf;
        bf.x = colsLDS[ka * TILE_N + nl];
        bf.y = colsLDS[(ka + 1) * TILE_N + nl];
        acc = __builtin_amdgcn_wmma_f32_16x16x4_f32(
            /*neg_a=*/false, a, /*neg_b=*/false, bf,
            /*c_mod=*/(short)0, acc, /*reuse_a=*/false, /*reuse_b=*/false);
    }

    // ---- Store D: VGPR r -> M = r (+8 for lanes 16-31), N = lane&15 ----
    float* ob = out + (size_t)b * O_SZ * HW + sp0 + nl;
    #pragma unroll
    for (int r = 0; r < 8; ++r) {
        int m = m0 + r + (hiHalf << 3);
        ob[(size_t)m * HW] = acc[r];
    }
}

// ---------------------------------------------------------------------------
extern "C" void kernel_launch(void* const* d_in, const int* in_sizes, int n_in,
                              void* d_out, int out_size, void* d_ws, size_t ws_size,
                              hipStream_t stream) {
    const float* x   = (const float*)d_in[0];   // (4,64,128,128)
    const float* wof = (const float*)d_in[1];   // (18,64,3,3)
    const float* wdf = (const float*)d_in[2];   // (64,64,3,3)
    float* out = (float*)d_out;                 // (4,64,128,128)
    float* off = (float*)d_ws;                  // (4,18,128,128) = 4.72 MB

    {   // offset conv
        int total = B_SZ * 18 * HW;
        dim3 grid((total + 255) / 256), block(256);
        offset_conv_kernel<<<grid, block, 0, stream>>>(x, wof, off);
    }
    {   // fused gather + WMMA GEMM
        int nTiles = (B_SZ * HW) / TILE_N;      // 4096
        deform_gemm_kernel<<<dim3(nTiles), dim3(128), 0, stream>>>(x, off, wdf, out);
    }
}